// TeacherEnsemble_3264175145229
// MI455X (gfx1250) — compile-verified
//
#include <hip/hip_runtime.h>

// ---------------- Soft-NMS (gaussian) for MI455X / gfx1250 -----------------
// Sequential algorithm -> single persistent workgroup on one WGP.
// State: scores in VGPRs (8 boxes/thread), boxes+labels staged in LDS via
// async global->LDS loads, one workgroup barrier per iteration.

#define NBOX 4096
#define NTH  512
#define PT   (NBOX / NTH)          // 8 boxes per thread
#define SCORE_TH 0.001f
#define NEG_INV_SIGMA (-2.0f)      // -1/SIGMA, SIGMA = 0.5 (gaussian method)

typedef int v4i __attribute__((vector_size(16)));           // 4 x i32
typedef __attribute__((address_space(1))) v4i as1_v4i;      // global
typedef __attribute__((address_space(3))) v4i as3_v4i;      // LDS

#if __has_builtin(__builtin_amdgcn_global_load_async_to_lds_b128)
#define HAVE_ASYNC_LDS 1
#else
#define HAVE_ASYNC_LDS 0
#endif

__device__ __forceinline__ void async_copy16(const void* g, void* l) {
#if HAVE_ASYNC_LDS
  // generic LDS pointer: low 32 bits are the LDS byte offset (as3 value)
  __builtin_amdgcn_global_load_async_to_lds_b128(
      (as1_v4i*)(unsigned long long)g,
      (as3_v4i*)(unsigned int)(unsigned long long)l,
      /*offset=*/0, /*cpol=*/0);
#else
  *reinterpret_cast<float4*>(l) = *reinterpret_cast<const float4*>(g);
#endif
}

__device__ __forceinline__ void async_copy_wait() {
#if HAVE_ASYNC_LDS
#if __has_builtin(__builtin_amdgcn_s_wait_asynccnt)
  __builtin_amdgcn_s_wait_asynccnt(0);
#else
  asm volatile("s_wait_asynccnt 0" ::: "memory");
#endif
#endif
}

__global__ __launch_bounds__(NTH, 1)
void softnms_persistent(const float* __restrict__ boxes,
                        const float* __restrict__ scores,
                        const int*   __restrict__ labels,
                        float* __restrict__ out) {
  __shared__ float lbox[NBOX * 4];   // 64 KB, read-only after init
  __shared__ int   llab[NBOX];       // 16 KB, read-only after init
  __shared__ float redv[64];         // double-buffered 32-slot reduce
  __shared__ int   redi[64];

  const int tid  = threadIdx.x;
  const int wv   = tid >> 5;         // wave id (0..15)
  const int lane = tid & 31;

  // sentinel-init both reduce buffers once; slots 16..31 never rewritten
  if (tid < 64) { redv[tid] = -3.0f; redi[tid] = 0; }

  // ---- stage boxes & labels into LDS (async path on CDNA5) ----
  for (int k = 0; k < PT; ++k) {
    int gi = tid + k * NTH;                       // strided ownership
    async_copy16(boxes + gi * 4, lbox + gi * 4);  // one box = 16B
  }
  for (int j = 0; j < NBOX / 4 / NTH; ++j) {      // 2 chunks of 4 ints
    int c = tid + j * NTH;
    async_copy16(labels + c * 4, llab + c * 4);
  }
  async_copy_wait();

  // ---- own boxes/scores/labels into registers (straight from global) ----
  float X1[PT], Y1[PT], X2[PT], Y2[PT], A[PT], S[PT];
  int   L[PT];
  for (int k = 0; k < PT; ++k) {
    int gi = tid + k * NTH;
    float4 b = reinterpret_cast<const float4*>(boxes)[gi];
    X1[k] = b.x; Y1[k] = b.y; X2[k] = b.z; Y2[k] = b.w;
    A[k]  = (b.z - b.x) * (b.w - b.y);
    S[k]  = scores[gi];                // active scores >= 0.05 > 0
    L[k]  = labels[gi];
  }
  __syncthreads();                     // LDS staging + redv init visible

  int buf = 0;
  for (int t = 0; t < NBOX; ++t) {
    // ---- local argmax over my 8 boxes (lowest index wins ties) ----
    float v = S[0]; int vi = tid;
    for (int k = 1; k < PT; ++k) {
      int gk = tid + k * NTH;
      bool better = (S[k] > v) || ((S[k] == v) && (gk < vi));
      v  = better ? S[k] : v;
      vi = better ? gk   : vi;
    }
    const float lv = v;                // my local max (for skip heuristic)

    // ---- wave32 butterfly reduce ----
    for (int off = 16; off >= 1; off >>= 1) {
      float ov = __shfl_xor(v,  off);
      int   oi = __shfl_xor(vi, off);
      bool better = (ov > v) || ((ov == v) && (oi < vi));
      v  = better ? ov : v;
      vi = better ? oi : vi;
    }
    if (lane == 0) { redv[buf * 32 + wv] = v; redi[buf * 32 + wv] = vi; }
    __syncthreads();                   // the ONLY barrier per iteration

    // every wave redundantly reduces the 32 slots -> no second barrier
    v  = redv[buf * 32 + lane];
    vi = redi[buf * 32 + lane];
    for (int off = 16; off >= 1; off >>= 1) {
      float ov = __shfl_xor(v,  off);
      int   oi = __shfl_xor(vi, off);
      bool better = (ov > v) || ((ov == v) && (oi < vi));
      v  = better ? ov : v;
      vi = better ? oi : vi;
    }
    buf ^= 1;

    const bool  valid = (v > 0.0f);    // inactive sentinel is -1
    const int   i     = vi;
    // broadcast-read winner's box from LDS (same address across lanes)
    const float bx1 = lbox[i * 4 + 0];
    const float by1 = lbox[i * 4 + 1];
    const float bx2 = lbox[i * 4 + 2];
    const float by2 = lbox[i * 4 + 3];
    const int   blab  = llab[i];
    const float barea = (bx2 - bx1) * (by2 - by1);

    // ---- emit outputs for step t ----
    if (tid == 0) {
      float zv = valid ? 1.0f : 0.0f;
      out[t * 4 + 0] = bx1 * zv;
      out[t * 4 + 1] = by1 * zv;
      out[t * 4 + 2] = bx2 * zv;
      out[t * 4 + 3] = by2 * zv;
      out[NBOX * 4 + t] = valid ? v : 0.0f;                 // score BEFORE decay
      out[NBOX * 5 + t] = valid ? (float)blab : -1.0f;      // label / padding
    }

    // ---- decay my scores (skip if all my boxes already inactive) ----
    if (lv > 0.0f) {
      for (int k = 0; k < PT; ++k) {
        float xx1 = fmaxf(bx1, X1[k]);
        float yy1 = fmaxf(by1, Y1[k]);
        float xx2 = fminf(bx2, X2[k]);
        float yy2 = fminf(by2, Y2[k]);
        float w   = fmaxf(xx2 - xx1, 0.0f);
        float h   = fmaxf(yy2 - yy1, 0.0f);
        float inter = w * h;
        float iou   = inter / (barea + A[k] - inter + 1e-8f);
        float decay = __expf(NEG_INV_SIGMA * iou * iou);
        float ns = (L[k] == blab) ? S[k] * decay : S[k];
        ns = (ns >= SCORE_TH) ? ns : -1.0f;                 // deactivate
        S[k] = ((tid + k * NTH) == i) ? -1.0f : ns;         // winner retired
      }
    }
  }
}

extern "C" void kernel_launch(void* const* d_in, const int* in_sizes, int n_in,
                              void* d_out, int out_size, void* d_ws, size_t ws_size,
                              hipStream_t stream) {
  (void)in_sizes; (void)n_in; (void)out_size; (void)d_ws; (void)ws_size;
  const float* boxes  = (const float*)d_in[0];   // [4096,4] f32
  const float* scores = (const float*)d_in[1];   // [4096]   f32
  const int*   labels = (const int*)  d_in[2];   // [4096]   i32
  float*       out    = (float*)d_out;           // 24576 f32
  softnms_persistent<<<1, NTH, 0, stream>>>(boxes, scores, labels, out);
}